// ModuleRenderScatterEX_12601434046909
// MI455X (gfx1250) — compile-verified
//
#include <hip/hip_runtime.h>
#include <stdint.h>

#define R      6
#define TS     16
#define PW     28
#define NPIX   (PW * PW)      // 784
#define K13    13
#define NK     (K13 * K13)    // 169
#define IMG_H  1024
#define IMG_W  1024
#define HW     (IMG_H * IMG_W)
#define NB     4
#define NCH    3

// --- gfx1250 async global->LDS path (probe via __has_builtin) -------------
#if defined(__has_builtin)
#  if __has_builtin(__builtin_amdgcn_global_load_async_to_lds_b32)
#    define HAVE_ASYNC_LDS 1
#  endif
#  if __has_builtin(__builtin_amdgcn_s_wait_asynccnt)
#    define HAVE_WAIT_ASYNC 1
#  endif
#endif

typedef __attribute__((address_space(1))) int* gint_p;
typedef __attribute__((address_space(3))) int* lint_p;

__device__ __forceinline__ void async_copy_b32(const void* g, void* l) {
#if defined(HAVE_ASYNC_LDS)
    __builtin_amdgcn_global_load_async_to_lds_b32(
        (gint_p)((void*)g),
        (lint_p)((void*)l),
        /*offset=*/0, /*cpol=*/0);
#else
    *(float*)l = *(const float*)g;   // fallback: normal load + ds_store
#endif
}

__device__ __forceinline__ void wait_async_lds() {
#if defined(HAVE_WAIT_ASYNC)
    __builtin_amdgcn_s_wait_asynccnt(0);
#elif defined(HAVE_ASYNC_LDS)
    asm volatile("s_wait_asynccnt 0" ::: "memory");
#endif
}

__global__ __launch_bounds__(256, 4)
void bokeh_render_kernel(const float* __restrict__ img,
                         const float* __restrict__ def,
                         float* __restrict__ out_bokeh,
                         float* __restrict__ out_dil)
{
    // staging (async landing zone)
    __shared__ float  s_def[NPIX];
    __shared__ float  s_im[3][NPIX];
    // packed hot-loop arrays
    __shared__ float2 s_di[NPIX];   // {defocus(-1e9 OOB sentinel), 1/(pi r^2+0.2)}
    __shared__ float4 s_pk[NPIX];   // {r, g, b, bits(int(defocus))}
    __shared__ float4 s_cst[NK];    // {dist_k, C_k, bits(deltaA*8), bits(deltaA*16)}

    const int tx  = threadIdx.x, ty = threadIdx.y;
    const int tid = ty * TS + tx;
    const int b   = blockIdx.z;

    // -------- per-offset constants (computed once per block, 169 lanes) ---
    if (tid < NK) {
        const int   dy  = tid / K13 - R;
        const int   dx  = tid % K13 - R;
        const float dyf = (float)dy, dxf = (float)dx;
        const float dist   = sqrtf(dyf * dyf + dxf * dxf);
        const float theta  = atan2f(dyf, dxf);
        const float sector = 6.2831853071795864769e0f / 10000.0f;  // 2*pi/POLY_SIDES
        const float a      = theta - 1.57079632679489662f;          // - INIT_ANGLE
        const float ang    = a - floorf(a / sector) * sector - 0.5f * sector; // python mod
        const float Cp     = cosf(0.5f * sector) / cosf(ang);
        const int   delta  = (-dy) * PW + (-dx);                    // gather: src = tgt - (dy,dx)
        s_cst[tid] = make_float4(dist, Cp,
                                 __int_as_float(delta * 8),
                                 __int_as_float(delta * 16));
    }

    // -------- async halo-tile fill: 28x28 defocus + 3 image channels ------
    const int gy0 = blockIdx.y * TS - R;
    const int gx0 = blockIdx.x * TS - R;
    const float* imgb = img + (size_t)b * NCH * HW;
    const float* defb = def + (size_t)b * HW;

    for (int i = tid; i < NPIX; i += 256) {
        const int iy = i / PW, ix = i - iy * PW;
        const int gy = gy0 + iy, gx = gx0 + ix;
        if ((unsigned)gy < (unsigned)IMG_H && (unsigned)gx < (unsigned)IMG_W) {
            const size_t gi = (size_t)gy * IMG_W + gx;
            async_copy_b32(defb + gi,            &s_def[i]);
            async_copy_b32(imgb + gi,            &s_im[0][i]);
            async_copy_b32(imgb + gi + HW,       &s_im[1][i]);
            async_copy_b32(imgb + gi + 2 * HW,   &s_im[2][i]);
        } else {
            // OOB sentinel: kills sigmoid (exp overflow -> rcp -> 0) and inside-test
            s_def[i]   = -1.0e9f;
            s_im[0][i] = 0.0f; s_im[1][i] = 0.0f; s_im[2][i] = 0.0f;
        }
    }
    wait_async_lds();
    __syncthreads();

    // -------- pack per-source-pixel derived values ------------------------
    for (int i = tid; i < NPIX; i += 256) {
        const float d   = s_def[i];
        const float inv = 1.0f / (3.14159274101257f * d * d + 0.2f);
        s_di[i] = make_float2(d, inv);
        s_pk[i] = make_float4(s_im[0][i], s_im[1][i], s_im[2][i],
                              __int_as_float((int)d));
    }
    __syncthreads();

    // -------- hot loop: 169 offsets, 2 trans + ~13 VALU + 3 DS each -------
    const char* pA0 = (const char*)&s_di[(ty + R) * PW + (tx + R)];
    const char* pB0 = (const char*)&s_pk[(ty + R) * PW + (tx + R)];

    float a0 = 0.0f, a1 = 0.0f, a2 = 0.0f, ws = 0.0f;
    int dil = (int)0x80000000;   // INT_MIN; offset (0,0) always covers init value

    #pragma unroll 13
    for (int k = 0; k < NK; ++k) {
        const float4 c  = s_cst[k];                                   // ds_load_b128 (broadcast)
        const float2 A  = *(const float2*)(pA0 + __float_as_int(c.z)); // ds_load_b64
        const float4 Bv = *(const float4*)(pB0 + __float_as_int(c.w)); // ds_load_b128
        const float t = fmaf(A.x, c.y, -c.x);                 // r_poly - dist
        // 0.5+0.5*tanh(4t) == sigmoid(8t) == 1/(1+exp(-8t)); exp(-8t)=2^(t*-8log2e)
        const float e = __builtin_amdgcn_exp2f(t * -11.5415603f);
        const float s = __builtin_amdgcn_rcpf(1.0f + e);
        const float w = s * A.y;
        ws += w;
        a0 = fmaf(w, Bv.x, a0);
        a1 = fmaf(w, Bv.y, a1);
        a2 = fmaf(w, Bv.z, a2);
        const int cand = (t >= 0.0f) ? __float_as_int(Bv.w) : (int)0x80000000;
        dil = max(dil, cand);
    }

    // -------- write out ---------------------------------------------------
    const int gy = blockIdx.y * TS + ty;
    const int gx = blockIdx.x * TS + tx;
    const size_t o = (size_t)b * NCH * HW + (size_t)gy * IMG_W + gx;
    out_bokeh[o]          = a0 / ws;
    out_bokeh[o + HW]     = a1 / ws;
    out_bokeh[o + 2 * HW] = a2 / ws;
    out_dil[(size_t)b * HW + (size_t)gy * IMG_W + gx] = (float)dil;
}

extern "C" void kernel_launch(void* const* d_in, const int* in_sizes, int n_in,
                              void* d_out, int out_size, void* d_ws, size_t ws_size,
                              hipStream_t stream) {
    (void)in_sizes; (void)n_in; (void)d_ws; (void)ws_size; (void)out_size;
    const float* img = (const float*)d_in[0];   // (4,3,1024,1024) f32
    const float* def = (const float*)d_in[1];   // (4,1,1024,1024) f32
    float* out      = (float*)d_out;                       // bokeh first
    float* out_dil  = out + (size_t)NB * NCH * HW;         // then dilate
    dim3 grid(IMG_W / TS, IMG_H / TS, NB);
    dim3 block(TS, TS, 1);
    hipLaunchKernelGGL(bokeh_render_kernel, grid, block, 0, stream,
                       img, def, out, out_dil);
}